// YoloLoss_27650999451734
// MI455X (gfx1250) — compile-verified
//
#include <hip/hip_runtime.h>
#include <math.h>

// ---------------------------------------------------------------------------
// YOLO rotated-box loss for MI455X (gfx1250, wave32).
//   obj : bce over all cells' channel 5 vs scatter-flag targets
//   box : per-target smooth-L1 (5 terms) at assigned level only (mask m==1)
//   cls : per-target  sum_c softplus+(logit_c) - logit_label  (one-hot BCE)
// cls class-sums are reduced with V_WMMA_F32_16X16X4_F32 (A = softplus values,
// B = ones, C accumulates) -> exact f32 row sums, 16 targets per wave.
// ---------------------------------------------------------------------------

typedef __attribute__((ext_vector_type(2))) float v2f;
typedef __attribute__((ext_vector_type(8))) float v8f;

#define NB 32
#define NT 100
#define NC 86
#define NCLS 80
#define CELLS_PER_B 8400            // 80*80 + 40*40 + 20*20
#define TOTAL_CELLS (NB * CELLS_PER_B)   // 268800  (divisible by 256)
#define FLAG_WORDS (TOTAL_CELLS / 32)    // 8400 u32 words in d_ws (33.6 KB)

#define L_BOX 0.05f
#define L_OBJ 1.0f
#define L_CLS 0.5f
#define PI_F 3.14159265358979323846f

__device__ __forceinline__ float softplus_pos(float x) {
    // max(x,0) + log1p(exp(-|x|))  == stable bce(x, t=0)
    return fmaxf(x, 0.f) + log1pf(expf(-fabsf(x)));
}
__device__ __forceinline__ float smooth_l1(float d) {
    float a = fabsf(d);
    return (a < 1.f) ? 0.5f * d * d : a - 0.5f;
}
__device__ __forceinline__ float clamp4(float x) {
    return fminf(fmaxf(x, -4.f), 4.f);
}

// ---- pass 0: zero flag bitmap and the scalar output --------------------------
__global__ void yolo_init_k(unsigned int* __restrict__ flags, float* __restrict__ out) {
    int id = blockIdx.x * blockDim.x + threadIdx.x;
    if (id < FLAG_WORDS) flags[id] = 0u;
    if (id == 0) out[0] = 0.f;
}

// ---- pass 1: per-target work (3200 targets, 16 per wave) ---------------------
__global__ void yolo_target_k(const float* __restrict__ p0, const float* __restrict__ p1,
                              const float* __restrict__ p2, const float* __restrict__ boxes,
                              const int* __restrict__ labels,
                              unsigned int* __restrict__ flags, float* __restrict__ out)
{
    const int lane  = threadIdx.x & 31;
    const int wid   = threadIdx.x >> 5;
    const int mrow  = lane & 15;      // A-matrix row (target within wave)
    const int khalf = lane >> 4;      // 0: K=0,1   1: K=2,3   (16x4 f32 A layout)
    const int t = (blockIdx.x * 8 + wid) * 16 + mrow;   // 0..3199, always in range
    const int b = t / NT;

    const float* bx = boxes + (size_t)t * 5;
    const float cx = bx[0], cy = bx[1], w = bx[2], h = bx[3], ang = bx[4];

    // level assignment: argmin |sqrt(max(w,1)*max(h,1)) - stride*4|, first-min ties
    const float gm = sqrtf(fmaxf(w, 1.f) * fmaxf(h, 1.f));
    const float d0 = fabsf(gm - 32.f), d1 = fabsf(gm - 64.f), d2 = fabsf(gm - 128.f);
    int lvl = 0; float best = d0;
    if (d1 < best) { lvl = 1; best = d1; }
    if (d2 < best) { lvl = 2; }

    const float s   = (lvl == 0) ? 8.f : ((lvl == 1) ? 16.f : 32.f);
    const int   Wg  = (lvl == 0) ? 80  : ((lvl == 1) ? 40 : 20);
    const int   lb0 = (lvl == 0) ? 0   : ((lvl == 1) ? 6400 : 8000);
    const float* pbase = (lvl == 0) ? p0 : ((lvl == 1) ? p1 : p2);

    const float gx = cx / s, gy = cy / s;
    const int gi = min(max((int)gx, 0), Wg - 1);
    const int gj = min(max((int)gy, 0), Wg - 1);
    const float* pr = pbase + (((size_t)b * Wg + gj) * Wg + gi) * NC;

    // one lane per target: objectness flag scatter (set semantics) + box loss
    if (khalf == 0) {
        const int cell = b * CELLS_PER_B + lb0 + gj * Wg + gi;
        atomicOr(&flags[cell >> 5], 1u << (cell & 31));

        const float tx = gx - (float)gi, ty = gy - (float)gj;
        const float gw = fmaxf(w / s, 1e-6f), gh = fmaxf(h / s, 1e-6f);
        const float tw = clamp4(logf(gw + 1e-6f)), th = clamp4(logf(gh + 1e-6f));
        const float pw = clamp4(pr[2]), ph = clamp4(pr[3]);
        float ad = fmodf(pr[4] - ang + PI_F, 2.f * PI_F);   // python-mod semantics
        ad = (ad < 0.f) ? ad + 2.f * PI_F : ad;
        ad -= PI_F;
        const float pt = smooth_l1(pr[0] - tx) + smooth_l1(pr[1] - ty)
                       + smooth_l1(pw - tw)    + smooth_l1(ph - th)
                       + smooth_l1(ad);
        atomicAdd(out, L_BOX * pt);
    }

    // every lane computes its target's label logit (lanes 16-31 mirror 0-15)
    const int lb = labels[t];
    const float xl = pr[6 + lb];

#if __has_builtin(__builtin_amdgcn_wmma_f32_16x16x4_f32)
    // cls row sums via f32 WMMA: A[m][k] = softplus+(logit), B = ones.
    // A 16x4 f32 layout: lanes 0-15 hold K=0 (v0), K=1 (v1); lanes 16-31 K=2,3.
    v8f acc = {};
    v2f onesb; onesb[0] = 1.f; onesb[1] = 1.f;
#pragma unroll
    for (int kb = 0; kb < NCLS; kb += 4) {
        const int c0 = 6 + kb + khalf * 2;
        v2f a;
        a[0] = softplus_pos(pr[c0]);
        a[1] = softplus_pos(pr[c0 + 1]);
        acc = __builtin_amdgcn_wmma_f32_16x16x4_f32(
            /*neg_a=*/false, a, /*neg_b=*/false, onesb,
            /*c_mod=*/(short)0, acc, /*reuse_a=*/false, /*reuse_b=*/false);
    }
    // D[m][n] identical across n. C/D layout: vgpr v -> M=v (lanes 0-15),
    // M=v+8 (lanes 16-31). Owners: lane m (0..7) -> target m via acc[lane];
    // lane 16+e (e=0..7) -> target e+8 via acc[e].
    const int  eidx  = (lane < 16) ? lane : (lane - 16);
    const bool owner = (lane < 8) || (lane >= 16 && lane < 24);
    float rs = acc[0];
    rs = (eidx == 1) ? acc[1] : rs;
    rs = (eidx == 2) ? acc[2] : rs;
    rs = (eidx == 3) ? acc[3] : rs;
    rs = (eidx == 4) ? acc[4] : rs;
    rs = (eidx == 5) ? acc[5] : rs;
    rs = (eidx == 6) ? acc[6] : rs;
    rs = (eidx == 7) ? acc[7] : rs;
    const int srcl = (lane < 16) ? lane : (lane - 8);  // lane holding target's xl
    const float xls = __shfl(xl, srcl, 32);
    if (owner) atomicAdd(out, L_CLS * (rs - xls));
#else
    // scalar fallback (host pass / builtin unavailable)
    if (khalf == 0) {
        float rs = 0.f;
        for (int c = 0; c < NCLS; ++c) rs += softplus_pos(pr[6 + c]);
        atomicAdd(out, L_CLS * (rs - xl));
    }
#endif
}

// ---- pass 2: objectness BCE over every cell + block reduction ----------------
__global__ void yolo_obj_k(const float* __restrict__ p0, const float* __restrict__ p1,
                           const float* __restrict__ p2,
                           const unsigned int* __restrict__ flags, float* __restrict__ out)
{
    const int id = blockIdx.x * blockDim.x + threadIdx.x;   // 0..268799 exact
    const int b = id / CELLS_PER_B;
    const int r = id - b * CELLS_PER_B;
    const float* p;
    size_t off;
    if (r < 6400)      { p = p0; off = (size_t)b * 6400 + r; }
    else if (r < 8000) { p = p1; off = (size_t)b * 1600 + (r - 6400); }
    else               { p = p2; off = (size_t)b * 400  + (r - 8000); }
    const float x  = p[off * NC + 5];
    const float tt = ((flags[id >> 5] >> (id & 31)) & 1u) ? 1.f : 0.f;
    float v = fmaxf(x, 0.f) - x * tt + log1pf(expf(-fabsf(x)));

    // wave32 shuffle reduction, then cross-wave via LDS, one atomic per block
#pragma unroll
    for (int o = 16; o > 0; o >>= 1) v += __shfl_down(v, o, 32);
    __shared__ float sm[8];
    const int lane = threadIdx.x & 31, wid = threadIdx.x >> 5;
    if (lane == 0) sm[wid] = v;
    __syncthreads();
    if (threadIdx.x == 0) {
        float sum = 0.f;
#pragma unroll
        for (int i = 0; i < 8; ++i) sum += sm[i];
        atomicAdd(out, L_OBJ * sum);
    }
}

extern "C" void kernel_launch(void* const* d_in, const int* in_sizes, int n_in,
                              void* d_out, int out_size, void* d_ws, size_t ws_size,
                              hipStream_t stream)
{
    (void)in_sizes; (void)n_in; (void)out_size; (void)ws_size;
    const float* p0    = (const float*)d_in[0];
    const float* p1    = (const float*)d_in[1];
    const float* p2    = (const float*)d_in[2];
    const float* boxes = (const float*)d_in[3];
    const int*   labels= (const int*)d_in[4];
    float* out = (float*)d_out;
    unsigned int* flags = (unsigned int*)d_ws;   // needs 33.6 KB of d_ws

    yolo_init_k<<<(FLAG_WORDS + 255) / 256, 256, 0, stream>>>(flags, out);
    yolo_target_k<<<25, 256, 0, stream>>>(p0, p1, p2, boxes, labels, flags, out);
    yolo_obj_k<<<TOTAL_CELLS / 256, 256, 0, stream>>>(p0, p1, p2, flags, out);
}